// GMM_7894149890423
// MI455X (gfx1250) — compile-verified
//
#include <hip/hip_runtime.h>
#include <math.h>

// ---------------------------------------------------------------------------
// GMM score kernel for MI455X (gfx1250), wave32.
//
// score(x) = -(sum_k c_k N_k(x) * inv_k (x-mu_k)) / (sum_k c_k N_k(x))
//
// Reformulated with y_k = inv_k x, b_k = inv_k mu_k, e_k = mu_k . b_k:
//   maha_k = x.y_k - 2 b_k.x + e_k
//   isft_k = y_k - b_k
//   w_k    = exp2( lw2_k - (0.5*log2 e)*maha_k ),  lw2_k = log2e*(log c_k - 0.5(logdet_k+4 log 2pi))
//
// All linear quantities (y_k, b_k.x) for a tile of 16 points are produced by
// TWO v_wmma_f32_16x16x4_f32 ops:
//   A1 (16x4): rows 4k+r = inv_k row r, k=0..3
//   A2 (16x4): rows 0-3 = inv_4 rows, rows 4..8 = b_0..b_4 (dot-product rows)
//   B  (4x16): column n = x of point n
// D layout (ISA 7.12.2): lane n (low) holds M=0..7, lane n+16 holds M=8..15 of
// column n -> a lane PAIR owns one point. Low lane processes components
// {0,1,4}, high lane {2,3} (slot 2 weight forced to 0 on the high half); three
// shfl_xor(16) exchange the b.x scalars and five reduce num/den across the pair.
// ---------------------------------------------------------------------------

typedef float v2f __attribute__((ext_vector_type(2)));
typedef float v8f __attribute__((ext_vector_type(8)));

#define GMM_K  5
#define WS_A1  0      // 16x4 floats, WMMA A layout source (row-major M x K)
#define WS_A2  64     // 16x4 floats
#define WS_LW  128    // 5 floats: log2-domain weights
#define WS_E   136    // 5 floats: mu^T inv mu
#define WS_B   144    // 5 x float4: b_k = inv_k mu_k
#define WS_TOT 176

__device__ __align__(16) float g_gmm_const[WS_TOT];

// ---- prep: invert 5 SPD 4x4 covariances (double), build constants ----------
__global__ void gmm_prep_kernel(const float* __restrict__ c,
                                const float* __restrict__ means,
                                const float* __restrict__ covs) {
  const int t = threadIdx.x;
  for (int i = t; i < WS_TOT; i += 32) g_gmm_const[i] = 0.0f;
  __syncthreads();
  if (t >= GMM_K) return;

  double m[16], inv[16];
#pragma unroll
  for (int i = 0; i < 16; ++i) m[i] = (double)covs[t * 16 + i];

  // 4x4 inverse via adjugate (layout-transpose-invariant; covs symmetric anyway)
  inv[0]  =  m[5]*m[10]*m[15] - m[5]*m[11]*m[14] - m[9]*m[6]*m[15] + m[9]*m[7]*m[14] + m[13]*m[6]*m[11] - m[13]*m[7]*m[10];
  inv[4]  = -m[4]*m[10]*m[15] + m[4]*m[11]*m[14] + m[8]*m[6]*m[15] - m[8]*m[7]*m[14] - m[12]*m[6]*m[11] + m[12]*m[7]*m[10];
  inv[8]  =  m[4]*m[9]*m[15]  - m[4]*m[11]*m[13] - m[8]*m[5]*m[15] + m[8]*m[7]*m[13] + m[12]*m[5]*m[11] - m[12]*m[7]*m[9];
  inv[12] = -m[4]*m[9]*m[14]  + m[4]*m[10]*m[13] + m[8]*m[5]*m[14] - m[8]*m[6]*m[13] - m[12]*m[5]*m[10] + m[12]*m[6]*m[9];
  inv[1]  = -m[1]*m[10]*m[15] + m[1]*m[11]*m[14] + m[9]*m[2]*m[15] - m[9]*m[3]*m[14] - m[13]*m[2]*m[11] + m[13]*m[3]*m[10];
  inv[5]  =  m[0]*m[10]*m[15] - m[0]*m[11]*m[14] - m[8]*m[2]*m[15] + m[8]*m[3]*m[14] + m[12]*m[2]*m[11] - m[12]*m[3]*m[10];
  inv[9]  = -m[0]*m[9]*m[15]  + m[0]*m[11]*m[13] + m[8]*m[1]*m[15] - m[8]*m[3]*m[13] - m[12]*m[1]*m[11] + m[12]*m[3]*m[9];
  inv[13] =  m[0]*m[9]*m[14]  - m[0]*m[10]*m[13] - m[8]*m[1]*m[14] + m[8]*m[2]*m[13] + m[12]*m[1]*m[10] - m[12]*m[2]*m[9];
  inv[2]  =  m[1]*m[6]*m[15]  - m[1]*m[7]*m[14]  - m[5]*m[2]*m[15] + m[5]*m[3]*m[14] + m[13]*m[2]*m[7]  - m[13]*m[3]*m[6];
  inv[6]  = -m[0]*m[6]*m[15]  + m[0]*m[7]*m[14]  + m[4]*m[2]*m[15] - m[4]*m[3]*m[14] - m[12]*m[2]*m[7]  + m[12]*m[3]*m[6];
  inv[10] =  m[0]*m[5]*m[15]  - m[0]*m[7]*m[13]  - m[4]*m[1]*m[15] + m[4]*m[3]*m[13] + m[12]*m[1]*m[7]  - m[12]*m[3]*m[5];
  inv[14] = -m[0]*m[5]*m[14]  + m[0]*m[6]*m[13]  + m[4]*m[1]*m[14] - m[4]*m[2]*m[13] - m[12]*m[1]*m[6]  + m[12]*m[2]*m[5];
  inv[3]  = -m[1]*m[6]*m[11]  + m[1]*m[7]*m[10]  + m[5]*m[2]*m[11] - m[5]*m[3]*m[10] - m[9]*m[2]*m[7]   + m[9]*m[3]*m[6];
  inv[7]  =  m[0]*m[6]*m[11]  - m[0]*m[7]*m[10]  - m[4]*m[2]*m[11] + m[4]*m[3]*m[10] + m[8]*m[2]*m[7]   - m[8]*m[3]*m[6];
  inv[11] = -m[0]*m[5]*m[11]  + m[0]*m[7]*m[9]   + m[4]*m[1]*m[11] - m[4]*m[3]*m[9]  - m[8]*m[1]*m[7]   + m[8]*m[3]*m[5];
  inv[15] =  m[0]*m[5]*m[10]  - m[0]*m[6]*m[9]   - m[4]*m[1]*m[10] + m[4]*m[2]*m[9]  + m[8]*m[1]*m[6]   - m[8]*m[2]*m[5];

  double det = m[0]*inv[0] + m[1]*inv[4] + m[2]*inv[8] + m[3]*inv[12];
  double invdet = 1.0 / det;
#pragma unroll
  for (int i = 0; i < 16; ++i) inv[i] *= invdet;

  double mu[4], b[4];
#pragma unroll
  for (int d = 0; d < 4; ++d) mu[d] = (double)means[t * 4 + d];
#pragma unroll
  for (int r = 0; r < 4; ++r) {
    double s = 0.0;
#pragma unroll
    for (int d = 0; d < 4; ++d) s += inv[r * 4 + d] * mu[d];
    b[r] = s;
  }
  double e = 0.0;
#pragma unroll
  for (int d = 0; d < 4; ++d) e += mu[d] * b[d];

  const double LOG_2PI = 1.8378770664093454836;
  const double LOG2E   = 1.4426950408889634074;
  double lw = log((double)c[t]) - 0.5 * (log(det) + 4.0 * LOG_2PI);

  if (t < 4) {
#pragma unroll
    for (int r = 0; r < 4; ++r)
#pragma unroll
      for (int d = 0; d < 4; ++d)
        g_gmm_const[WS_A1 + (4 * t + r) * 4 + d] = (float)inv[r * 4 + d];
  } else {
#pragma unroll
    for (int r = 0; r < 4; ++r)
#pragma unroll
      for (int d = 0; d < 4; ++d)
        g_gmm_const[WS_A2 + r * 4 + d] = (float)inv[r * 4 + d];
  }
#pragma unroll
  for (int d = 0; d < 4; ++d) g_gmm_const[WS_A2 + (4 + t) * 4 + d] = (float)b[d];
  g_gmm_const[WS_LW + t] = (float)(lw * LOG2E);
  g_gmm_const[WS_E + t]  = (float)e;
#pragma unroll
  for (int d = 0; d < 4; ++d) g_gmm_const[WS_B + t * 4 + d] = (float)b[d];
}

// ---- main: WMMA-tiled score, 32 points per wave per iteration --------------
__global__ __launch_bounds__(256) void gmm_score_kernel(
    const float4* __restrict__ x4, float4* __restrict__ out4, int npts) {
  const unsigned lane = threadIdx.x & 31u;
  const unsigned r    = lane & 15u;     // M (A) / N (B,D) index within half
  const bool     hi   = lane >= 16u;
  const unsigned koff = hi ? 2u : 0u;   // A/B: low lanes K=0,1 ; high lanes K=2,3
  const float* gc = g_gmm_const;

  // A operands in WMMA 32-bit 16x4 layout: V0 = K=0|K=2, V1 = K=1|K=3, M=lane%16
  v2f a1, a2;
  a1.x = gc[WS_A1 + r * 4 + koff];
  a1.y = gc[WS_A1 + r * 4 + koff + 1];
  a2.x = gc[WS_A2 + r * 4 + koff];
  a2.y = gc[WS_A2 + r * 4 + koff + 1];

  // per-lane component slots: low = {0,1,4}, high = {2,3,(4 with w=0)}
  const int k0 = hi ? 2 : 0;
  const int k1 = hi ? 3 : 1;
  const float lw_s0 = gc[WS_LW + k0];
  const float lw_s1 = gc[WS_LW + k1];
  const float lw_s2 = hi ? -__builtin_inff() : gc[WS_LW + 4];
  const float e_s0  = gc[WS_E + k0];
  const float e_s1  = gc[WS_E + k1];
  const float e_s2  = gc[WS_E + 4];
  const float4* bp  = (const float4*)(gc + WS_B);
  const float4 b_s0 = bp[k0];
  const float4 b_s1 = bp[k1];
  const float4 b_s2 = bp[4];

  const unsigned wavesPerBlock = blockDim.x >> 5;
  const unsigned wid     = blockIdx.x * wavesPerBlock + (threadIdx.x >> 5);
  const unsigned wstride = gridDim.x * wavesPerBlock;
  const unsigned ntiles  = (unsigned)((npts + 31) >> 5);
  const float HL2E = 0.72134752044448170367f;  // 0.5 * log2(e)

  for (unsigned tile = wid; tile < ntiles; tile += wstride) {
    const int base = (int)(tile << 5);
#pragma unroll
    for (int g = 0; g < 2; ++g) {
      const int p  = base + g * 16 + (int)r;
      const int pc = p < npts ? p : npts - 1;   // clamp: keep EXEC all-ones for WMMA
      const float4 xv = x4[pc];

      // B operand (4x16): column = point, low lanes carry dims 0/1, high 2/3
      v2f bm;
      bm.x = hi ? xv.z : xv.x;
      bm.y = hi ? xv.w : xv.y;

      v8f zero = {0.f, 0.f, 0.f, 0.f, 0.f, 0.f, 0.f, 0.f};
      // D1 col n = [y0(4), y1(4) | y2(4), y3(4)], D2 col n = [y4(4), d0..d3 | d4, 0...]
      v8f d1 = __builtin_amdgcn_wmma_f32_16x16x4_f32(false, a1, false, bm,
                                                     (short)0, zero, false, false);
      v8f d2 = __builtin_amdgcn_wmma_f32_16x16x4_f32(false, a2, false, bm,
                                                     (short)0, zero, false, false);

      // exchange b.x scalars across the lane pair
      const float t0 = __shfl_xor(d2[0], 16, 32);  // low gets d4
      const float t6 = __shfl_xor(d2[6], 16, 32);  // high gets d2
      const float t7 = __shfl_xor(d2[7], 16, 32);  // high gets d3
      const float d_s0 = hi ? t6 : d2[4];
      const float d_s1 = hi ? t7 : d2[5];
      const float d_s2 = t0;

      float nx = 0.f, ny = 0.f, nz = 0.f, nw = 0.f, den = 0.f;
      {  // slot 0: comp 0 (low) / comp 2 (high), y = d1[0..3]
        float maha = xv.x*d1[0] + xv.y*d1[1] + xv.z*d1[2] + xv.w*d1[3]
                     - 2.0f * d_s0 + e_s0;
        float w = exp2f(lw_s0 - HL2E * maha);
        nx += w * (d1[0] - b_s0.x); ny += w * (d1[1] - b_s0.y);
        nz += w * (d1[2] - b_s0.z); nw += w * (d1[3] - b_s0.w);
        den += w;
      }
      {  // slot 1: comp 1 (low) / comp 3 (high), y = d1[4..7]
        float maha = xv.x*d1[4] + xv.y*d1[5] + xv.z*d1[6] + xv.w*d1[7]
                     - 2.0f * d_s1 + e_s1;
        float w = exp2f(lw_s1 - HL2E * maha);
        nx += w * (d1[4] - b_s1.x); ny += w * (d1[5] - b_s1.y);
        nz += w * (d1[6] - b_s1.z); nw += w * (d1[7] - b_s1.w);
        den += w;
      }
      {  // slot 2: comp 4 on low lanes only (lw = -inf -> w = 0 on high lanes)
        float maha = xv.x*d2[0] + xv.y*d2[1] + xv.z*d2[2] + xv.w*d2[3]
                     - 2.0f * d_s2 + e_s2;
        float w = exp2f(lw_s2 - HL2E * maha);
        nx += w * (d2[0] - b_s2.x); ny += w * (d2[1] - b_s2.y);
        nz += w * (d2[2] - b_s2.z); nw += w * (d2[3] - b_s2.w);
        den += w;
      }

      // reduce num/den across the lane pair
      nx  += __shfl_xor(nx, 16, 32);
      ny  += __shfl_xor(ny, 16, 32);
      nz  += __shfl_xor(nz, 16, 32);
      nw  += __shfl_xor(nw, 16, 32);
      den += __shfl_xor(den, 16, 32);

      const float s = -1.0f / den;
      if (!hi && p < npts) out4[p] = make_float4(nx * s, ny * s, nz * s, nw * s);
    }
  }
}

extern "C" void kernel_launch(void* const* d_in, const int* in_sizes, int n_in,
                              void* d_out, int out_size, void* d_ws, size_t ws_size,
                              hipStream_t stream) {
  const float* x     = (const float*)d_in[0];
  const float* c     = (const float*)d_in[1];
  const float* means = (const float*)d_in[2];
  const float* covs  = (const float*)d_in[3];
  float* out = (float*)d_out;
  const int npts = in_sizes[0] / 4;

  gmm_prep_kernel<<<1, 32, 0, stream>>>(c, means, covs);

  const int tiles  = (npts + 31) / 32;
  int blocks = (tiles + 7) / 8;           // 8 waves per 256-thread block
  if (blocks > 4096) blocks = 4096;       // grid-stride covers the rest
  if (blocks < 1) blocks = 1;
  gmm_score_kernel<<<blocks, 256, 0, stream>>>(
      (const float4*)x, (float4*)out, npts);
}